// GraphSAGE_Full_17016660426789
// MI455X (gfx1250) — compile-verified
//
#include <hip/hip_runtime.h>
#include <hip/hip_bf16.h>

typedef __attribute__((ext_vector_type(2))) float v2f;
typedef __attribute__((ext_vector_type(8))) float v8f;

// ---------------------------------------------------------------------------
// Zero-fill workspace buffer (agg / deg) -- graph-capture-safe (no memsetAsync)
// ---------------------------------------------------------------------------
__global__ void zero_kernel(float* __restrict__ p, int n) {
    int i = blockIdx.x * blockDim.x + threadIdx.x;
    if (i < n) p[i] = 0.0f;
}

// ---------------------------------------------------------------------------
// In-degree accumulation: deg[dst[e]] += 1
// ---------------------------------------------------------------------------
__global__ void deg_kernel(const int* __restrict__ dst, float* __restrict__ deg, int nEdges) {
    int e = blockIdx.x * blockDim.x + threadIdx.x;
    if (e < nEdges) atomicAdd(&deg[dst[e]], 1.0f);
}

// deg -> inv_deg = 1/max(deg,1) in place
__global__ void invdeg_kernel(float* __restrict__ deg, int n) {
    int i = blockIdx.x * blockDim.x + threadIdx.x;
    if (i < n) deg[i] = 1.0f / fmaxf(deg[i], 1.0f);
}

// ---------------------------------------------------------------------------
// Edge scatter-add: one wave per edge, 32 lanes x float2 = 64 features.
// agg[dst[e], :] += feat[src[e], :]
// ---------------------------------------------------------------------------
__global__ __launch_bounds__(256) void edge_scatter(const float* __restrict__ feat,
                                                    const int* __restrict__ src,
                                                    const int* __restrict__ dst,
                                                    float* __restrict__ agg,
                                                    int nEdges) {
    int wave = (blockIdx.x * blockDim.x + threadIdx.x) >> 5;
    if (wave >= nEdges) return;
    int lane = threadIdx.x & 31;
    int s = src[wave];
    int d = dst[wave];
    const float2* frow = (const float2*)(feat + (size_t)s * 64);
    float2 v = frow[lane];
    float* drow = agg + (size_t)d * 64 + lane * 2;
    atomicAdd(drow + 0, v.x);
    atomicAdd(drow + 1, v.y);
}

// ---------------------------------------------------------------------------
// SAGE layer GEMM via V_WMMA_F32_16X16X4_F32 (full f32 precision):
//   OUT[16 rows x 64 cols] = H @ Wself + (AGG * inv_deg) @ Wneigh + b  (opt ReLU)
// One wave per 16-row tile; 4 accumulators cover all 64 output columns.
// A-frag f32 16x4: lane<16 -> row=lane, K=kb..kb+1 ; lane>=16 -> row=lane-16, K=kb+2..kb+3
// B-frag f32 4x16: lane<16 -> col=lane, K=kb..kb+1 ; lane>=16 -> col=lane-16, K=kb+2..kb+3
// C/D  f32 16x16: VGPR r -> M = r + (lane>=16 ? 8 : 0), N = lane & 15
// ---------------------------------------------------------------------------
__global__ __launch_bounds__(256) void sage_gemm_wmma(const float* __restrict__ H,
                                                      const float* __restrict__ AGG,
                                                      const float* __restrict__ INVD,
                                                      const float* __restrict__ Wself,
                                                      const float* __restrict__ Wneigh,
                                                      const float* __restrict__ bias,
                                                      float* __restrict__ OUT,
                                                      int nTiles, int doRelu) {
    int wave = (blockIdx.x * blockDim.x + threadIdx.x) >> 5;
    if (wave >= nTiles) return;               // wave-uniform exit: EXEC stays all-1s for WMMA
    int lane = threadIdx.x & 31;
    int l15  = lane & 15;
    int hi   = lane >> 4;                     // 0 or 1
    int rowbase = wave * 16;

    float invd = INVD[rowbase + l15];         // A-frag row for this lane is always l15

    // init accumulators with broadcast bias: C[r] = b[col], col = t*16 + l15
    v8f c[4];
#pragma unroll
    for (int t = 0; t < 4; ++t) {
        float b = bias[t * 16 + l15];
#pragma unroll
        for (int r = 0; r < 8; ++r) c[t][r] = b;
    }

    const float* hrow = H   + (size_t)(rowbase + l15) * 64;
    const float* arow = AGG + (size_t)(rowbase + l15) * 64;

#pragma unroll 4
    for (int kb = 0; kb < 64; kb += 4) {
        int k0 = kb + hi * 2;                 // this lane's K pair base
        v2f aS, aN;
        aS[0] = hrow[k0];
        aS[1] = hrow[k0 + 1];
        aN[0] = arow[k0] * invd;
        aN[1] = arow[k0 + 1] * invd;
#pragma unroll
        for (int t = 0; t < 4; ++t) {
            int col = t * 16 + l15;
            v2f bS, bN;
            bS[0] = Wself[(size_t)k0 * 64 + col];
            bS[1] = Wself[(size_t)(k0 + 1) * 64 + col];
            bN[0] = Wneigh[(size_t)k0 * 64 + col];
            bN[1] = Wneigh[(size_t)(k0 + 1) * 64 + col];
            // 8 args: (neg_a, A, neg_b, B, c_mod, C, reuse_a, reuse_b)
            c[t] = __builtin_amdgcn_wmma_f32_16x16x4_f32(
                false, aS, false, bS, (short)0, c[t], false, false);
            c[t] = __builtin_amdgcn_wmma_f32_16x16x4_f32(
                false, aN, false, bN, (short)0, c[t], false, false);
        }
    }

    // store D: row = rowbase + r + hi*8 , col = t*16 + l15
#pragma unroll
    for (int t = 0; t < 4; ++t) {
#pragma unroll
        for (int r = 0; r < 8; ++r) {
            float v = c[t][r];
            if (doRelu) v = fmaxf(v, 0.0f);
            int row = rowbase + r + hi * 8;
            int col = t * 16 + l15;
            OUT[(size_t)row * 64 + col] = v;
        }
    }
}

// ---------------------------------------------------------------------------
// Launch: deg -> invdeg -> scatter(x) -> GEMM0(relu) -> rezero -> scatter(h) -> GEMM1
// ---------------------------------------------------------------------------
extern "C" void kernel_launch(void* const* d_in, const int* in_sizes, int n_in,
                              void* d_out, int out_size, void* d_ws, size_t ws_size,
                              hipStream_t stream) {
    const float* x       = (const float*)d_in[0];
    const int*   src     = (const int*)d_in[1];
    const int*   dst     = (const int*)d_in[2];
    const float* Wself0  = (const float*)d_in[3];
    const float* Wneigh0 = (const float*)d_in[4];
    const float* b0      = (const float*)d_in[5];
    const float* Wself1  = (const float*)d_in[6];
    const float* Wneigh1 = (const float*)d_in[7];
    const float* b1      = (const float*)d_in[8];
    float* out = (float*)d_out;

    const int D       = 64;
    const int nNodes  = in_sizes[0] / D;     // 50000
    const int nEdges  = in_sizes[1];         // 800000
    const int nFeat   = nNodes * D;          // 3,200,000
    const int nTiles  = (nNodes + 15) / 16;  // 3125 (exact)

    // workspace layout (floats): [inv_deg: nNodes][agg: nNodes*D][hidden: nNodes*D]
    float* ws      = (float*)d_ws;
    float* invdeg  = ws;
    float* agg     = ws + nNodes;
    float* hidden  = agg + (size_t)nFeat;

    const int TB = 256;
    dim3 blk(TB);

    // 1) zero deg + agg
    zero_kernel<<<dim3((nNodes + TB - 1) / TB), blk, 0, stream>>>(invdeg, nNodes);
    zero_kernel<<<dim3((nFeat + TB - 1) / TB), blk, 0, stream>>>(agg, nFeat);

    // 2) degree, then inv_deg = 1/max(deg,1)
    deg_kernel<<<dim3((nEdges + TB - 1) / TB), blk, 0, stream>>>(dst, invdeg, nEdges);
    invdeg_kernel<<<dim3((nNodes + TB - 1) / TB), blk, 0, stream>>>(invdeg, nNodes);

    // 3) layer 0 aggregation + GEMM (internal ReLU; outer ReLU idempotent)
    int scatterBlocks = (nEdges * 32 + TB - 1) / TB;   // one wave per edge
    edge_scatter<<<dim3(scatterBlocks), blk, 0, stream>>>(x, src, dst, agg, nEdges);
    int gemmBlocks = (nTiles * 32 + TB - 1) / TB;      // one wave per 16-row tile
    sage_gemm_wmma<<<dim3(gemmBlocks), blk, 0, stream>>>(
        x, agg, invdeg, Wself0, Wneigh0, b0, hidden, nTiles, /*relu=*/1);

    // 4) layer 1 aggregation + GEMM (no activation)
    zero_kernel<<<dim3((nFeat + TB - 1) / TB), blk, 0, stream>>>(agg, nFeat);
    edge_scatter<<<dim3(scatterBlocks), blk, 0, stream>>>(hidden, src, dst, agg, nEdges);
    sage_gemm_wmma<<<dim3(gemmBlocks), blk, 0, stream>>>(
        hidden, agg, invdeg, Wself1, Wneigh1, b1, out, nTiles, /*relu=*/0);
}